// CSRGridBuilder_91079076479408
// MI455X (gfx1250) — compile-verified
//
#include <hip/hip_runtime.h>

#ifndef __has_builtin
#define __has_builtin(x) 0
#endif

#if defined(__HIP_DEVICE_COMPILE__) && __has_builtin(__builtin_amdgcn_tensor_load_to_lds)
#define USE_TDM 1
#else
#define USE_TDM 0
#endif

typedef __attribute__((ext_vector_type(4))) unsigned int v4u;
typedef __attribute__((ext_vector_type(8))) int v8i;
typedef __attribute__((ext_vector_type(4))) int v4i;
typedef __attribute__((ext_vector_type(2))) float v2f;
typedef __attribute__((ext_vector_type(8))) float v8f;

#define TILE 4096           // elements (uint2 = 8B) per sort tile -> 32KB LDS
#define SENT 0xFFFFFFFFu

// ---------------- morton helpers ----------------
__device__ __forceinline__ unsigned expand10(unsigned x) {
  x &= 1023u;
  x = (x | (x << 16)) & 0x030000FFu;
  x = (x | (x << 8))  & 0x0300F00Fu;
  x = (x | (x << 4))  & 0x030C30C3u;
  x = (x | (x << 2))  & 0x09249249u;
  return x;
}

__device__ __forceinline__ void sphere_grid(const float* __restrict__ mn,
                                            const float* __restrict__ mx,
                                            const float* __restrict__ gmin,
                                            float vs, int G, int i,
                                            int gl[3], int e[3]) {
#pragma unroll
  for (int c = 0; c < 3; ++c) {
    int lo = (int)floorf((mn[3 * i + c] - gmin[c]) / vs);
    int hi = (int)floorf((mx[3 * i + c] - gmin[c]) / vs);
    lo = min(max(lo, 0), G - 1);
    hi = min(max(hi, 0), G - 1);
    gl[c] = lo;
    e[c] = hi - lo + 1;
  }
}

// ---------------- WMMA-based 256-element inclusive scan (wave32) ----------------
// Values viewed as a 16x16 f32 matrix, linear index = m*16 + n.
//   R = X * U   (U upper-triangular ones)  -> row-wise inclusive prefix
//   D = Lstrict * Y + R, Y[r,n]=rowsum[r]  -> add totals of preceding rows
// 8 chained V_WMMA_F32_16X16X4_F32, no LDS, no barriers. EXEC must be all-1s:
// callers are 32-thread blocks with no divergence around this call.
// Result: lane l (n=l%16, h=l>=16), vgpr v holds inclusive[(v+8*h)*16 + n].
template <typename F>
__device__ __forceinline__ v8f wmma_scan256(int lane, F load) {
  const int n = lane & 15;
  const int h = (lane >> 4) & 1;
  // X in A-layout: lane holds row (l%16), elements k = 4c + v + 2h
  float xa[8];
#pragma unroll
  for (int c = 0; c < 4; ++c)
#pragma unroll
    for (int v = 0; v < 2; ++v) xa[c * 2 + v] = load(n * 16 + 4 * c + v + 2 * h);
  v8f acc = {};
#pragma unroll
  for (int c = 0; c < 4; ++c) {
    v2f a = {xa[c * 2 + 0], xa[c * 2 + 1]};
    v2f bu = {(4 * c + 0 + 2 * h <= n) ? 1.0f : 0.0f,
              (4 * c + 1 + 2 * h <= n) ? 1.0f : 0.0f};
    acc = __builtin_amdgcn_wmma_f32_16x16x4_f32(false, a, false, bu, (short)0, acc,
                                                false, false);
  }
  // rowsum[m] = R[m,15]: m<8 -> vgpr m lane 15; m>=8 -> vgpr m-8 lane 31
  float rs[16];
#pragma unroll
  for (int r = 0; r < 8; ++r) {
    rs[r] = __shfl(acc[r], 15, 32);
    rs[r + 8] = __shfl(acc[r], 31, 32);
  }
#pragma unroll
  for (int c = 0; c < 4; ++c) {
    v2f a = {(4 * c + 0 + 2 * h < n) ? 1.0f : 0.0f,   // strict lower-triangular
             (4 * c + 1 + 2 * h < n) ? 1.0f : 0.0f};
    v2f b = {rs[4 * c + 0 + 2 * h], rs[4 * c + 1 + 2 * h]};
    acc = __builtin_amdgcn_wmma_f32_16x16x4_f32(false, a, false, b, (short)0, acc,
                                                false, false);
  }
  return acc;
}

// ---------------- per-sphere setup ----------------
__global__ void setup_kernel(const float* __restrict__ mn, const float* __restrict__ mx,
                             const float* __restrict__ gmin, const float* __restrict__ vsz,
                             const int* __restrict__ gsz, int T, int M,
                             int* __restrict__ sphCounts, int* __restrict__ out_os) {
  int i = blockIdx.x * blockDim.x + threadIdx.x;
  if (i >= M) return;
  float vs = vsz[0];
  int G = gsz[0];
  int gl[3], e[3];
  sphere_grid(mn, mx, gmin, vs, G, i, gl, e);
  int nv = e[0] * e[1] * e[2];
  int os = (nv > T) ? 1 : 0;
  out_os[i] = os;
  sphCounts[i] = os ? 0 : nv;
}

// ---------------- offsets scan (WMMA wave scan; counts <= 64 so f32 exact) ----------------
__global__ void offsets_partial_kernel(const int* __restrict__ cnts, int* __restrict__ out_off,
                                       int* __restrict__ blockSums, int M) {
  int lane = threadIdx.x;
  int base = blockIdx.x * 256;
  v8f inc = wmma_scan256(lane, [&](int idx) -> float {
    int i = base + idx;
    return (i < M) ? (float)cnts[i] : 0.0f;
  });
  int n = lane & 15, h = (lane >> 4) & 1;
#pragma unroll
  for (int v = 0; v < 8; ++v) {
    int i = base + (v + 8 * h) * 16 + n;
    if (i < M) out_off[i + 1] = (int)inc[v];
  }
  float tot = __shfl(inc[7], 31, 32);
  if (lane == 0) blockSums[blockIdx.x] = (int)tot;
}

// integer Hillis-Steele scan (exclusive, in place) — kept for the digit-base scan
// whose values can exceed 2^24 (f32-exact range).
__global__ void scan_ex_kernel(int* __restrict__ data, int n) {
  __shared__ int sm[256];
  int carry = 0;
  for (int base = 0; base < n; base += 256) {
    int i = base + threadIdx.x;
    int v = (i < n) ? data[i] : 0;
    sm[threadIdx.x] = v;
    __syncthreads();
    for (int off = 1; off < 256; off <<= 1) {
      int t = (threadIdx.x >= off) ? sm[threadIdx.x - off] : 0;
      __syncthreads();
      sm[threadIdx.x] += t;
      __syncthreads();
    }
    if (i < n) data[i] = sm[threadIdx.x] - v + carry;
    carry += sm[255];
    __syncthreads();
  }
}

// WMMA exclusive scan in place, single 32-thread block (values small -> f32 exact)
__global__ void scan_ex_small_kernel(int* __restrict__ data, int nElems) {
  int lane = threadIdx.x;
  int carry = 0;
  for (int base = 0; base < nElems; base += 256) {
    v8f inc = wmma_scan256(lane, [&](int idx) -> float {
      int i = base + idx;
      return (i < nElems) ? (float)data[i] : 0.0f;
    });
    int n = lane & 15, h = (lane >> 4) & 1;
#pragma unroll
    for (int v = 0; v < 8; ++v) {
      int i = base + (v + 8 * h) * 16 + n;
      if (i < nElems) data[i] = (int)inc[v] - data[i] + carry;
    }
    carry += (int)__shfl(inc[7], 31, 32);
  }
}

__global__ void offsets_addback_kernel(int* __restrict__ out_off, const int* __restrict__ blockSums,
                                       int M, int* __restrict__ out_tot) {
  int i = blockIdx.x * 256 + threadIdx.x;
  if (i == 0) out_off[0] = 0;
  if (i < M) {
    int val = out_off[i + 1] + blockSums[blockIdx.x];
    out_off[i + 1] = val;
    if (i == M - 1) out_tot[0] = val;
  }
}

// ---------------- pair expansion ----------------
__global__ void expand_kernel(const float* __restrict__ mn, const float* __restrict__ mx,
                              const float* __restrict__ gmin, const float* __restrict__ vsz,
                              const int* __restrict__ gsz, int T, int M, long long MT,
                              uint2* __restrict__ A) {
  long long gid = (long long)blockIdx.x * blockDim.x + threadIdx.x;
  if (gid >= MT) return;
  int i = (int)(gid / T);
  int k = (int)(gid % T);
  float vs = vsz[0];
  int G = gsz[0];
  int gl[3], e[3];
  sphere_grid(mn, mx, gmin, vs, G, i, gl, e);
  int nv = e[0] * e[1] * e[2];
  int cnt = (nv > T) ? 0 : nv;
  unsigned key;
  int sid;
  if (k < cnt) {
    int ez = e[2], ey = e[1];
    int dz = k % ez;
    int dy = (k / ez) % ey;
    int dx = k / (ez * ey);
    unsigned gx = (unsigned)min(gl[0] + dx, 1023);
    unsigned gy = (unsigned)min(gl[1] + dy, 1023);
    unsigned gz = (unsigned)min(gl[2] + dz, 1023);
    key = (expand10(gx) << 2) | (expand10(gy) << 1) | expand10(gz);
    sid = i;
  } else {
    key = SENT;
    sid = -1;
  }
  A[gid] = make_uint2(key, (unsigned)sid);
}

// ---------------- radix sort: histogram (TDM tile staging) ----------------
__global__ void hist_kernel(const uint2* __restrict__ src, int* __restrict__ counts,
                            int totalPairs, int numBlocks, int shift) {
  __shared__ uint2 tile[TILE];
  __shared__ int hist[256];
  int b = blockIdx.x, t = threadIdx.x;
  int tileStart = b * TILE;
  int n = min(TILE, totalPairs - tileStart);
  hist[t] = 0;
#if USE_TDM
  if (t < 32) {  // wave 0 only issues the DMA (EXEC is ignored by TENSOR ops)
    unsigned long long ga = (unsigned long long)(const void*)(src + tileStart);
    unsigned lds = (unsigned)(unsigned long long)(void*)&tile[0];
    // D# group0: count=1(valid), lds_addr, global_addr[56:0], type=2 ("image")
    v4u g0 = {1u, lds, (unsigned)ga,
              (unsigned)((ga >> 32) & 0x1FFFFFFull) | (2u << 30)};
    unsigned td0 = (unsigned)n;  // valid elements; OOB reads return zero
    // D# group1: data_size=3 (8B), tensor_dim0=n, tensor_dim1=1, tile_dim0=TILE,
    //            tensor_dim0_stride=n, everything else zero
    v8i g1 = {(int)(3u << 16),
              (int)((td0 & 0xFFFFu) << 16),
              (int)(((td0 >> 16) & 0xFFFFu) | (1u << 16)),
              (int)(((unsigned)TILE) << 16),
              0,
              (int)td0,
              0,
              0};
    v4i gz4 = {0, 0, 0, 0};
    v8i gz8 = {0, 0, 0, 0, 0, 0, 0, 0};
    __builtin_amdgcn_tensor_load_to_lds(g0, g1, gz4, gz4, gz8, 0);
    __builtin_amdgcn_s_wait_tensorcnt(0);
    asm volatile("" ::: "memory");
  }
  __syncthreads();
#else
  for (int i = t; i < n; i += 256) tile[i] = src[tileStart + i];
  __syncthreads();
#endif
  for (int i = t; i < n; i += 256) {
    unsigned d = (tile[i].x >> shift) & 255u;
    atomicAdd(&hist[d], 1);
  }
  __syncthreads();
  counts[(size_t)t * numBlocks + b] = hist[t];  // digit-major layout
}

// ---------------- radix sort: per-digit scan over blocks (WMMA wave scan) ----------------
// per-block digit counts <= 4096, chunk totals <= 2^20 -> f32 exact.
__global__ void scan_blocks_kernel(int* __restrict__ counts, int* __restrict__ digitTotal,
                                   int numBlocks) {
  int lane = threadIdx.x;
  int d = blockIdx.x;
  int* row = counts + (size_t)d * numBlocks;
  int carry = 0;
  for (int base = 0; base < numBlocks; base += 256) {
    v8f inc = wmma_scan256(lane, [&](int idx) -> float {
      int i = base + idx;
      return (i < numBlocks) ? (float)row[i] : 0.0f;
    });
    int n = lane & 15, h = (lane >> 4) & 1;
#pragma unroll
    for (int v = 0; v < 8; ++v) {
      int i = base + (v + 8 * h) * 16 + n;
      if (i < numBlocks) row[i] = (int)inc[v] - row[i] + carry;  // exclusive
    }
    carry += (int)__shfl(inc[7], 31, 32);
  }
  if (lane == 0) digitTotal[d] = carry;
}

__global__ void add_base_kernel(int* __restrict__ counts, const int* __restrict__ digitBase,
                                int numBlocks, int total) {
  int i = blockIdx.x * 256 + threadIdx.x;
  if (i < total) counts[i] += digitBase[i / numBlocks];
}

// ---------------- radix sort: stable scatter ----------------
// 128 threads, 32 consecutive elements each; u16 rank matrix 128x256 = 64KB LDS.
template <bool FINAL>
__global__ void scatter_kernel(const uint2* __restrict__ src, uint2* __restrict__ dst,
                               long long* __restrict__ out_m, int* __restrict__ out_sid,
                               const int* __restrict__ ofs, int totalPairs, int numBlocks,
                               int shift) {
  __shared__ unsigned short rank[128][256];
  int b = blockIdx.x, t = threadIdx.x;
  int tileStart = b * TILE;
  int n = min(TILE, totalPairs - tileStart);
  int* rz = (int*)rank;
  for (int i = t; i < 128 * 256 / 2; i += 128) rz[i] = 0;
  __syncthreads();
  const uint2* tsrc = src + tileStart;
  int base = t * 32;
  // phase 1: count own chunk's digits
  for (int j = 0; j < 32; ++j) {
    int li = base + j;
    if (li < n) {
      unsigned d = (tsrc[li].x >> shift) & 255u;
      rank[t][d]++;
    }
  }
  __syncthreads();
  // phase 2: per-digit exclusive scan across threads (stable, thread order = memory order)
  for (int d = t; d < 256; d += 128) {
    unsigned s = 0;
    for (int tt = 0; tt < 128; ++tt) {
      unsigned tmp = rank[tt][d];
      rank[tt][d] = (unsigned short)s;
      s += tmp;
    }
  }
  __syncthreads();
  // phase 3: scatter (global offsets already include digitBase)
  for (int j = 0; j < 32; ++j) {
    int li = base + j;
    if (li < n) {
      uint2 kv = tsrc[li];
      unsigned d = (kv.x >> shift) & 255u;
      int r = rank[t][d];
      rank[t][d] = (unsigned short)(r + 1);
      int idx = ofs[(size_t)d * numBlocks + b] + r;
      if (FINAL) {
        out_m[idx] = (kv.x == SENT) ? (1LL << 40) : (long long)kv.x;
        out_sid[idx] = (int)kv.y;
      } else {
        dst[idx] = kv;
      }
    }
  }
}

// ---------------- host orchestration ----------------
extern "C" void kernel_launch(void* const* d_in, const int* in_sizes, int n_in,
                              void* d_out, int out_size, void* d_ws, size_t ws_size,
                              hipStream_t stream) {
  (void)n_in;
  (void)ws_size;
  const float* mn = (const float*)d_in[0];
  const float* mx = (const float*)d_in[1];
  const float* gmin = (const float*)d_in[2];
  const float* vsz = (const float*)d_in[3];
  const int* gsz = (const int*)d_in[4];
  // d_in[5] = oversized_threshold on device; recover T from out_size instead:
  int M = in_sizes[0] / 3;
  long long T = ((long long)out_size - 2LL * M - 2LL) / (2LL * M);
  if (T < 1) T = 1;
  long long MT = (long long)M * T;
  int totalPairs = (int)MT;
  int numBlocks = (int)((MT + TILE - 1) / TILE);

  // output layout (return order): int64 morton[MT], int32 sid[MT],
  // int32 offsets[M+1], int32 is_oversized[M], int32 total[1]
  char* o = (char*)d_out;
  long long* out_m = (long long*)o;
  int* out_sid = (int*)(o + (size_t)MT * 8);
  int* out_off = out_sid + MT;
  int* out_os = out_off + (M + 1);
  int* out_tot = out_os + M;

  // workspace layout
  char* w = (char*)d_ws;
  auto alloc = [&](size_t bytes) {
    char* p = w;
    w += (bytes + 255) & ~(size_t)255;
    return p;
  };
  uint2* A = (uint2*)alloc((size_t)MT * 8);
  uint2* B = (uint2*)alloc((size_t)MT * 8);
  int* counts = (int*)alloc((size_t)256 * numBlocks * 4);
  int* digitBase = (int*)alloc(256 * 4);
  int* sphCounts = (int*)alloc((size_t)M * 4);
  int nb1 = (M + 255) / 256;
  int* blockSums = (int*)alloc((size_t)nb1 * 4);

  // 1) per-sphere counts + is_oversized
  setup_kernel<<<(M + 255) / 256, 256, 0, stream>>>(mn, mx, gmin, vsz, gsz, (int)T, M,
                                                    sphCounts, out_os);
  // 2) CSR offsets scan (WMMA wave scans)
  offsets_partial_kernel<<<nb1, 32, 0, stream>>>(sphCounts, out_off, blockSums, M);
  scan_ex_small_kernel<<<1, 32, 0, stream>>>(blockSums, nb1);
  offsets_addback_kernel<<<nb1, 256, 0, stream>>>(out_off, blockSums, M, out_tot);
  // 3) expand to 32-bit (key,sid) pairs
  expand_kernel<<<(unsigned)((MT + 255) / 256), 256, 0, stream>>>(mn, mx, gmin, vsz, gsz,
                                                                  (int)T, M, MT, A);
  // 4) 4-pass LSD radix sort (stable); last pass unpacks straight into d_out
  const uint2* srcs[4] = {A, B, A, B};
  uint2* dsts[3] = {B, A, B};
  for (int p = 0; p < 4; ++p) {
    int shift = 8 * p;
    hist_kernel<<<numBlocks, 256, 0, stream>>>(srcs[p], counts, totalPairs, numBlocks, shift);
    scan_blocks_kernel<<<256, 32, 0, stream>>>(counts, digitBase, numBlocks);
    scan_ex_kernel<<<1, 256, 0, stream>>>(digitBase, 256);  // values can exceed 2^24 -> int
    add_base_kernel<<<(unsigned)(((size_t)256 * numBlocks + 255) / 256), 256, 0, stream>>>(
        counts, digitBase, numBlocks, 256 * numBlocks);
    if (p < 3) {
      scatter_kernel<false><<<numBlocks, 128, 0, stream>>>(
          srcs[p], dsts[p], nullptr, nullptr, counts, totalPairs, numBlocks, shift);
    } else {
      scatter_kernel<true><<<numBlocks, 128, 0, stream>>>(
          srcs[p], nullptr, out_m, out_sid, counts, totalPairs, numBlocks, shift);
    }
  }
}